// FlexibleCMC_23278722745130
// MI455X (gfx1250) — compile-verified
//
#include <hip/hip_runtime.h>

typedef __attribute__((ext_vector_type(2))) float v2f;
typedef __attribute__((ext_vector_type(8))) float v8f;

#define NODES   128
#define STATES  8
#define ROWB    16      // batch rows per block
#define THREADS 256     // 8 wave32s

// ---------------------------------------------------------------------------
// Prep: build B-operand matrix for WMMA from node_connectivity.
//   node_input[b,i] = sum_j m[b,j] * C[i,j],  C = conn with zero diagonal
// GEMM form: D = M (BxK) * Bt (KxN) with Bt[k][n] = conn[n*128+k]*(n!=k).
// Packed for the 32-bit 4x16 WMMA B layout so each lane does ONE float2 load:
//   Bp[(k>>1)*256 + n*2 + (k&1)] = Bt[k][n]
// ---------------------------------------------------------------------------
__global__ void cmc_prep_B(const float* __restrict__ conn, float* __restrict__ Bp) {
    int t = blockIdx.x * 256 + threadIdx.x;        // 0 .. 16383
    if (t >= NODES * NODES) return;
    int k = t >> 7;          // K index (source node j)
    int n = t & 127;         // N index (dest node i)
    float v = (n == k) ? 0.0f : conn[n * NODES + k];
    Bp[(k >> 1) * (2 * NODES) + n * 2 + (k & 1)] = v;
}

__device__ __forceinline__ float sigm_c(float z) {   // sigmoid(z) - 0.5
    return 1.0f / (1.0f + __expf(-z)) - 0.5f;
}
__device__ __forceinline__ float clip100(float v) {
    return fminf(fmaxf(v, -100.0f), 100.0f);
}

// ---------------------------------------------------------------------------
// Fused kernel: clip -> LDS, mean -> m, WMMA node_input, elementwise -> out
// ---------------------------------------------------------------------------
__global__ __launch_bounds__(THREADS)
void cmc_kernel(const float* __restrict__ x,
                const float* __restrict__ Bp,
                const float* __restrict__ G,
                const float* __restrict__ T,
                const float* __restrict__ adapt,
                const float* __restrict__ R,
                float* __restrict__ out) {
    __shared__ float xs[ROWB * NODES * STATES];   // 16 x 1024 floats (64 KB), clipped
    __shared__ float m[ROWB * NODES];             // 16 x 128 floats (8 KB)

    const int tid = threadIdx.x;
    const int b0  = blockIdx.x * ROWB;

    // ---- Phase 1: stage clipped x tile into LDS (read x from HBM exactly once)
    {
        const float4* __restrict__ xin = (const float4*)(x + (size_t)b0 * NODES * STATES);
        float4* xl = (float4*)xs;
        #pragma unroll
        for (int i = 0; i < (ROWB * NODES * STATES / 4) / THREADS; ++i) {  // 16 iters
            int idx = tid + THREADS * i;
            float4 v = xin[idx];
            v.x = clip100(v.x); v.y = clip100(v.y);
            v.z = clip100(v.z); v.w = clip100(v.w);
            xl[idx] = v;
        }
    }
    __syncthreads();

    // ---- Phase 2: m[row][n] = mean over 8 states (of clipped x)
    #pragma unroll
    for (int i = 0; i < (ROWB * NODES) / THREADS; ++i) {  // 8 iters
        int idx = tid + THREADS * i;                      // idx = row*128 + n
        const float4* p = (const float4*)&xs[(idx >> 7) * (NODES * STATES) + (idx & 127) * STATES];
        float4 a = p[0], b = p[1];
        m[idx] = (a.x + a.y + a.z + a.w + b.x + b.y + b.z + b.w) * 0.125f;
    }
    __syncthreads();

    // ---- Phase 3: node_input tile via V_WMMA_F32_16X16X4_F32 (fp32, matches ref)
    const int wave = tid >> 5;        // 0..7 -> which 16-node column tile
    const int lane = tid & 31;
    const int lo   = lane & 15;
    const int hi   = lane >> 4;
    const int n    = wave * 16 + lo;  // this lane's node column (B/D operand N)

    v8f acc = {};                           // C = 0
    const float* mrow = &m[lo * NODES];     // A row (M = lo) for this lane
    #pragma unroll
    for (int k0 = 0; k0 < NODES; k0 += 4) {
        const int kb = k0 + 2 * hi;         // ISA A/B layout: hi half-wave holds K+2,K+3
        v2f a = *(const v2f*)&mrow[kb];                                   // ds_load_b64
        v2f b = *(const v2f*)&Bp[(k0 / 2 + hi) * (2 * NODES) + n * 2];    // coalesced b64 (L2)
        acc = __builtin_amdgcn_wmma_f32_16x16x4_f32(
                  /*neg_a=*/false, a, /*neg_b=*/false, b,
                  /*c_mod=*/(short)0, acc, /*reuse_a=*/false, /*reuse_b=*/false);
    }

    // ---- Phase 4: per-node params (fixed per lane), then elementwise over 8 rows
    const float Rn = R[n];
    const float g0  = G[n*12+0],  g1  = G[n*12+1],  g2  = G[n*12+2],  g3 = G[n*12+3];
    const float g4  = G[n*12+4],  g5  = G[n*12+5],  g6  = G[n*12+6],  g7 = G[n*12+7];
    const float g8  = G[n*12+8],  g9  = G[n*12+9],  g10 = G[n*12+10], g11 = G[n*12+11];
    const float rt0 = 1.0f / T[n*4+0];
    const float rt1 = 1.0f / T[n*4+1];
    const float rt2 = 1.0f / T[n*4+2];
    const float rt3 = 1.0f / T[n*4+3];
    const float scale = 0.1f / (1.0f + __expf(-adapt[n]));

    #pragma unroll
    for (int r = 0; r < 8; ++r) {
        const int row = r + 8 * hi;                 // D layout: reg r, hi half -> M = r+8
        const float* xp = &xs[row * (NODES * STATES) + n * STATES];
        const float x0 = xp[0], x1 = xp[1], x2 = xp[2], x3 = xp[3];
        const float x4 = xp[4], x5 = xp[5], x6 = xp[6], x7 = xp[7];
        const float ni = acc[r];

        const float S0 = sigm_c(Rn * x0);
        const float S2 = sigm_c(Rn * x2);
        const float S4 = sigm_c(Rn * x4);
        const float S6 = sigm_c(Rn * x6);

        const float u1 = -g0 * S0 - g2 * S4 - g1 * S2 + ni;
        const float f1 = scale * (u1 - 2.0f * x1 - x0 * rt0) * rt0;
        const float u3 =  g7 * S0 - g6 * S2 - g11 * S4 + ni;
        const float f3 = scale * (u3 - 2.0f * x3 - x2 * rt1) * rt1;
        const float u5 =  g4 * S0 + g5 * S6 - g3 * S4 + g10 * S2 + ni;
        const float f5 = scale * (u5 - 2.0f * x5 - x4 * rt2) * rt2;
        const float u7 = -g9 * S6 - g8 * S4 + ni;
        const float f7 = scale * (u7 - 2.0f * x7 - x6 * rt3) * rt3;

        float4* op = (float4*)(out + (size_t)(b0 + row) * (NODES * STATES) + n * STATES);
        op[0] = make_float4(scale * x1, f1, scale * x3, f3);
        op[1] = make_float4(scale * x5, f5, scale * x7, f7);
    }
}

// ---------------------------------------------------------------------------
extern "C" void kernel_launch(void* const* d_in, const int* in_sizes, int n_in,
                              void* d_out, int out_size, void* d_ws, size_t ws_size,
                              hipStream_t stream) {
    (void)n_in; (void)out_size; (void)ws_size;
    // setup_inputs order: t, x, node_connectivity, G, T, adaptation, R
    const float* x     = (const float*)d_in[1];
    const float* conn  = (const float*)d_in[2];
    const float* G     = (const float*)d_in[3];
    const float* T     = (const float*)d_in[4];
    const float* adapt = (const float*)d_in[5];
    const float* R     = (const float*)d_in[6];
    float*       out   = (float*)d_out;
    float*       Bp    = (float*)d_ws;   // 128*128 floats = 64 KB scratch

    const int B = in_sizes[1] / (NODES * STATES);   // 32768

    cmc_prep_B<<<(NODES * NODES + 255) / 256, 256, 0, stream>>>(conn, Bp);
    cmc_kernel<<<B / ROWB, THREADS, 0, stream>>>(x, Bp, G, T, adapt, R, out);
}